// GlobalNodeInductiveCriticNetwork_29729763623417
// MI455X (gfx1250) — compile-verified
//
#include <hip/hip_runtime.h>
#include <cfloat>

typedef __attribute__((ext_vector_type(2))) float v2f;
typedef __attribute__((ext_vector_type(8))) float v8f;

#if __has_builtin(__builtin_amdgcn_global_load_async_to_lds_b32) && \
    __has_builtin(__builtin_amdgcn_s_wait_asynccnt)
#define USE_ASYNC_LDS 1
typedef __attribute__((address_space(1))) int async_gint;
typedef __attribute__((address_space(3))) int async_lint;
#else
#define USE_ASYNC_LDS 0
#endif

#define KMAX 512                      // largest K used (g0W/g1W: 512)
#define LDS_PITCH_PAD 2               // bank-conflict padding per column

// ---------------------------------------------------------------------------
// WMMA fp32 GEMM:  C[M,N] = act( gather(A)[M,K] @ B[K,N] + bias[N] )
// Block = 256 threads = 8 waves. Block owns one 16-wide column panel of C and
// 8 consecutive 16-row tiles (M must be a multiple of 128 -- true for all
// call sites: 3328, 51200, 256).  B[K x 16] is staged once into LDS
// (column-major, pitch K+2 for bank spread), shared by all 8 waves.
// Each wave then runs K/4 V_WMMA_F32_16X16X4_F32 steps.
// A layout: lanes 0-15 -> K{k,k+1}, lanes 16-31 -> K{k+2,k+3} (float2 load).
// C layout: VGPR r -> M = r (+8 for lanes 16-31), N = lane&15.
// ---------------------------------------------------------------------------
__global__ void gemm_wmma_f32(const float* __restrict__ A,
                              const float* __restrict__ Bm,
                              const float* __restrict__ bias,
                              float* __restrict__ C,
                              const int* __restrict__ rowIdx,
                              int K, int N, int relu)
{
  __shared__ float Blds[16 * (KMAX + LDS_PITCH_PAD)];
  const int tid     = threadIdx.x;
  const int lane    = tid & 31;
  const int waveId  = tid >> 5;
  const int colBase = blockIdx.x << 4;
  const int rowBase = ((blockIdx.y << 3) + waveId) << 4;
  const int pitch   = K + LDS_PITCH_PAD;

  // ---- stage B[K x 16] panel into LDS (column-major) ----
  for (int e = tid; e < (K << 4); e += 256) {
    int k = e >> 4, c = e & 15;
    const float* gp = Bm + (size_t)k * N + colBase + c;
    float* lp = &Blds[c * pitch + k];
#if USE_ASYNC_LDS
    __builtin_amdgcn_global_load_async_to_lds_b32(
        (async_gint*)gp, (async_lint*)lp, 0, 0);
#else
    *lp = *gp;
#endif
  }
#if USE_ASYNC_LDS
  __builtin_amdgcn_s_wait_asynccnt(0);
#endif
  __syncthreads();

  const int sub = lane & 15;
  const int hi  = lane >> 4;                       // 0 -> K{0,1}, 1 -> K{2,3}
  int arow = rowBase + sub;
  if (rowIdx) arow = rowIdx[arow];
  const float2* Ap = (const float2*)(A + (size_t)arow * K + (hi << 1)); // 8B aligned
  const float*  Bl = Blds + sub * pitch + (hi << 1);

  v8f acc = {0.f, 0.f, 0.f, 0.f, 0.f, 0.f, 0.f, 0.f};
  for (int k = 0; k < K; k += 4) {
    float2 av = Ap[k >> 1];                        // A[arow, k+hi*2 .. +1]
    v2f a, b;
    a.x = av.x;  a.y = av.y;
    b.x = Bl[k]; b.y = Bl[k + 1];                  // ds_load (8B aligned pair)
    acc = __builtin_amdgcn_wmma_f32_16x16x4_f32(false, a, false, b,
                                                (short)0, acc, false, false);
  }

  const int cCol  = colBase + sub;
  const int cRow0 = rowBase + (hi << 3);
  const float bb = bias ? bias[cCol] : 0.f;
#pragma unroll
  for (int r = 0; r < 8; ++r) {
    float v = acc[r] + bb;
    if (relu) v = fmaxf(v, 0.f);
    C[(size_t)(cRow0 + r) * N + cCol] = v;
  }
}

// ---------------- elementwise / reduction helpers --------------------------

__global__ void fill_f32(float* __restrict__ p, size_t n, float v) {
  size_t i = (size_t)blockIdx.x * blockDim.x + threadIdx.x;
  if (i < n) p[i] = v;
}

__global__ void deg_accum(const long long* __restrict__ dst,
                          float* __restrict__ deg, long long E) {
  size_t i = (size_t)blockIdx.x * blockDim.x + threadIdx.x;
  if (i < (size_t)E) atomicAdd(deg + dst[i], 1.0f);
}

__global__ void rsqrt_k(const float* __restrict__ deg,
                        float* __restrict__ dinv, int n) {
  int i = blockIdx.x * blockDim.x + threadIdx.x;
  if (i < n) dinv[i] = rsqrtf(deg[i]);
}

// one thread per (edge, 4-feature group): float4 gather + 4 atomic max
__global__ void scatter_max_edges_v4(const long long* __restrict__ src,
                                     const long long* __restrict__ dst,
                                     const float* __restrict__ dinv,
                                     const float* __restrict__ xw,
                                     float* __restrict__ agg,
                                     long long E, int F)
{
  size_t gid = (size_t)blockIdx.x * blockDim.x + threadIdx.x;
  size_t groups = (size_t)(F >> 2);
  size_t total = (size_t)E * groups;
  if (gid >= total) return;
  size_t e = gid / groups;
  int f4 = (int)(gid - e * groups) << 2;
  long long s = src[e], d = dst[e];
  float norm = dinv[s] * dinv[d];
  const float4 v = *(const float4*)(xw + (size_t)s * F + f4);
  float* out = agg + (size_t)d * F + f4;
  __builtin_prefetch(out, 1, 0);                   // global_prefetch_b8
  atomicMax(out + 0, norm * v.x);
  atomicMax(out + 1, norm * v.y);
  atomicMax(out + 2, norm * v.z);
  atomicMax(out + 3, norm * v.w);
}

__global__ void scatter_max_self_v4(const float* __restrict__ dinv,
                                    const float* __restrict__ xw,
                                    float* __restrict__ agg,
                                    int N, int F)
{
  size_t gid = (size_t)blockIdx.x * blockDim.x + threadIdx.x;
  size_t groups = (size_t)(F >> 2);
  size_t total = (size_t)N * groups;
  if (gid >= total) return;
  size_t i = gid / groups;
  int f4 = (int)(gid - i * groups) << 2;
  float di = dinv[i];
  float norm = di * di;
  const float4 v = *(const float4*)(xw + (size_t)i * F + f4);
  float* out = agg + (size_t)i * F + f4;
  atomicMax(out + 0, norm * v.x);
  atomicMax(out + 1, norm * v.y);
  atomicMax(out + 2, norm * v.z);
  atomicMax(out + 3, norm * v.w);
}

__global__ void bias_relu(const float* __restrict__ in,
                          const float* __restrict__ b,
                          float* __restrict__ out, size_t n, int F)
{
  size_t gid = (size_t)blockIdx.x * blockDim.x + threadIdx.x;
  if (gid >= n) return;
  out[gid] = fmaxf(in[gid] + b[gid % (size_t)F], 0.f);
}

// one wave32 per row, in-place softmax
__global__ void softmax_rows(float* __restrict__ X, int rows, int cols) {
  int wave = (int)((blockIdx.x * (size_t)blockDim.x + threadIdx.x) >> 5);
  int lane = threadIdx.x & 31;
  if (wave >= rows) return;
  float* row = X + (size_t)wave * cols;
  float m = -FLT_MAX;
  for (int c = lane; c < cols; c += 32) m = fmaxf(m, row[c]);
  for (int o = 16; o > 0; o >>= 1) m = fmaxf(m, __shfl_xor(m, o, 32));
  float s = 0.f;
  for (int c = lane; c < cols; c += 32) { float e = __expf(row[c] - m); row[c] = e; s += e; }
  for (int o = 16; o > 0; o >>= 1) s += __shfl_xor(s, o, 32);
  float inv = 1.f / s;
  for (int c = lane; c < cols; c += 32) row[c] *= inv;
}

// out[b,f] = sum_h att[b*NH+h, f] * feat[b*NH+h, f]
__global__ void attn_reduce(const float* __restrict__ att,
                            const float* __restrict__ feat,
                            float* __restrict__ out, int Bn, int NH, int F)
{
  int gid = blockIdx.x * blockDim.x + threadIdx.x;
  if (gid >= Bn * F) return;
  int b = gid / F, f = gid - b * F;
  float s = 0.f;
  for (int h = 0; h < NH; ++h) {
    size_t i = ((size_t)(b * NH + h)) * F + f;
    s += att[i] * feat[i];
  }
  out[gid] = s;
}

__global__ void concat2(const float* __restrict__ A, int Fa,
                        const float* __restrict__ Bv, int Fb,
                        float* __restrict__ out, int rows)
{
  int F = Fa + Fb;
  int gid = blockIdx.x * blockDim.x + threadIdx.x;
  if (gid >= rows * F) return;
  int r = gid / F, c = gid - r * F;
  out[gid] = (c < Fa) ? A[(size_t)r * Fa + c] : Bv[(size_t)r * Fb + (c - Fa)];
}

__global__ void add_inplace(float* __restrict__ a,
                            const float* __restrict__ b, size_t n) {
  size_t i = (size_t)blockIdx.x * blockDim.x + threadIdx.x;
  if (i < n) a[i] += b[i];
}

__global__ void head_out(const float* __restrict__ h, const float* __restrict__ w,
                         const float* __restrict__ b, float* __restrict__ out,
                         int Bn, int K)
{
  int i = blockIdx.x * blockDim.x + threadIdx.x;
  if (i >= Bn) return;
  float s = b[0];
  for (int k = 0; k < K; ++k) s += h[(size_t)i * K + k] * w[k];
  out[i] = s;
}

// ---------------------------------------------------------------------------

static inline unsigned cdiv(size_t a, unsigned b) { return (unsigned)((a + b - 1) / b); }

static void launch_gemm(const float* A, const float* B, const float* bias, float* C,
                        const int* rowIdx, int M, int K, int N, int relu,
                        hipStream_t stream)
{
  dim3 grid(N >> 4, (unsigned)(M >> 7));           // M multiple of 128 at all sites
  gemm_wmma_f32<<<grid, 256, 0, stream>>>(A, B, bias, C, rowIdx, K, N, relu);
}

extern "C" void kernel_launch(void* const* d_in, const int* in_sizes, int n_in,
                              void* d_out, int out_size, void* d_ws, size_t ws_size,
                              hipStream_t stream)
{
  const float*     x    = (const float*)d_in[0];
  const long long* ei   = (const long long*)d_in[1];   // int64 [2,E]
  const int*       hidx = (const int*)d_in[2];         // [3328]
  const float *W1 = (const float*)d_in[3],  *b1 = (const float*)d_in[4];
  const float *W2 = (const float*)d_in[5],  *b2 = (const float*)d_in[6];
  const float *a0W = (const float*)d_in[7],  *a0b = (const float*)d_in[8];
  const float *f0W = (const float*)d_in[9],  *f0b = (const float*)d_in[10];
  const float *g0W = (const float*)d_in[11], *g0b = (const float*)d_in[12];
  const float *a1W = (const float*)d_in[13], *a1b = (const float*)d_in[14];
  const float *f1W = (const float*)d_in[15], *f1b = (const float*)d_in[16];
  const float *g1W = (const float*)d_in[17], *g1b = (const float*)d_in[18];
  const float *a2W = (const float*)d_in[19], *a2b = (const float*)d_in[20];
  const float *f2W = (const float*)d_in[21], *f2b = (const float*)d_in[22];
  const float *g2W = (const float*)d_in[23], *g2b = (const float*)d_in[24];
  const float *o1W = (const float*)d_in[25], *o1b = (const float*)d_in[26];
  const float *o2W = (const float*)d_in[27], *o2b = (const float*)d_in[28];

  const int IN = 64, H1 = 256, H2 = 64, Bn = 256, NH = 13;
  const int N = in_sizes[0] / IN;                 // 51200
  const long long E = in_sizes[1] / 2;            // 409600
  const int NHB = in_sizes[2];                    // 3328 = Bn*NH
  const long long *esrc = ei, *edst = ei + E;

  // ---- workspace layout (floats) ----
  float* ws = (float*)d_ws;
  size_t NF1 = (size_t)N * H1;
  float* xw  = ws;                 // [N,H1]  (reused as xw2 [N,H2])
  float* agg = ws + NF1;           // [N,H1]  (reused as agg2)
  float* x1  = ws + 2 * NF1;       // [N,H1]  (x2 [N,H2] overwrites after last use)
  float* p = ws + 3 * NF1;
  float* deg  = p; p += N;
  float* dinv = p; p += N;
  float* att  = p; p += (size_t)NHB * H1;
  float* feat = p; p += (size_t)NHB * H1;
  float* outb = p; p += (size_t)Bn * H1;
  float* cat  = p; p += (size_t)Bn * (2 * H1);
  float* g    = p; p += (size_t)Bn * H1;
  float* gtmp = p; p += (size_t)Bn * H1;
  float* hbuf = p; p += (size_t)Bn * H2;
  float* xw2 = xw; float* agg2 = agg; float* x2 = x1;

  // ---- degree / normalization (shared by both GCN layers) ----
  fill_f32<<<cdiv(N, 256), 256, 0, stream>>>(deg, N, 1.0f);           // self-loops
  deg_accum<<<cdiv((size_t)E, 256), 256, 0, stream>>>(edst, deg, E);
  rsqrt_k<<<cdiv(N, 256), 256, 0, stream>>>(deg, dinv, N);

  // ---- attention stage 0 (v = x[hidx], g starts at 0) ----
  launch_gemm(x, a0W, a0b, att,  hidx, NHB, IN, H1, 0, stream);
  softmax_rows<<<cdiv((size_t)NHB * 32, 256), 256, 0, stream>>>(att, NHB, H1);
  launch_gemm(x, f0W, f0b, feat, hidx, NHB, IN, H1, 0, stream);
  attn_reduce<<<cdiv((size_t)Bn * H1, 256), 256, 0, stream>>>(att, feat, outb, Bn, NH, H1);
  fill_f32<<<cdiv((size_t)Bn * H1, 256), 256, 0, stream>>>(g, (size_t)Bn * H1, 0.f);
  concat2<<<cdiv((size_t)Bn * 2 * H1, 256), 256, 0, stream>>>(outb, H1, g, H1, cat, Bn);
  launch_gemm(cat, g0W, g0b, gtmp, nullptr, Bn, 2 * H1, H1, 0, stream);
  add_inplace<<<cdiv((size_t)Bn * H1, 256), 256, 0, stream>>>(g, gtmp, (size_t)Bn * H1);

  // ---- GCN layer 1: x1 = relu(max-agg(norm * (x@W1)) + b1) ----
  launch_gemm(x, W1, nullptr, xw, nullptr, N, IN, H1, 0, stream);
  fill_f32<<<cdiv(NF1, 256), 256, 0, stream>>>(agg, NF1, -FLT_MAX);
  scatter_max_edges_v4<<<cdiv((size_t)E * (H1 >> 2), 256), 256, 0, stream>>>(
      esrc, edst, dinv, xw, agg, E, H1);
  scatter_max_self_v4<<<cdiv((size_t)N * (H1 >> 2), 256), 256, 0, stream>>>(
      dinv, xw, agg, N, H1);
  bias_relu<<<cdiv(NF1, 256), 256, 0, stream>>>(agg, b1, x1, NF1, H1);

  // ---- attention stage 1 ----
  launch_gemm(x1, a1W, a1b, att,  hidx, NHB, H1, H1, 0, stream);
  softmax_rows<<<cdiv((size_t)NHB * 32, 256), 256, 0, stream>>>(att, NHB, H1);
  launch_gemm(x1, f1W, f1b, feat, hidx, NHB, H1, H1, 0, stream);
  attn_reduce<<<cdiv((size_t)Bn * H1, 256), 256, 0, stream>>>(att, feat, outb, Bn, NH, H1);
  concat2<<<cdiv((size_t)Bn * 2 * H1, 256), 256, 0, stream>>>(outb, H1, g, H1, cat, Bn);
  launch_gemm(cat, g1W, g1b, gtmp, nullptr, Bn, 2 * H1, H1, 0, stream);
  add_inplace<<<cdiv((size_t)Bn * H1, 256), 256, 0, stream>>>(g, gtmp, (size_t)Bn * H1);

  // ---- GCN layer 2: x2 = relu(max-agg(norm * (x1@W2)) + b2) ----
  size_t NF2 = (size_t)N * H2;
  launch_gemm(x1, W2, nullptr, xw2, nullptr, N, H1, H2, 0, stream);
  fill_f32<<<cdiv(NF2, 256), 256, 0, stream>>>(agg2, NF2, -FLT_MAX);
  scatter_max_edges_v4<<<cdiv((size_t)E * (H2 >> 2), 256), 256, 0, stream>>>(
      esrc, edst, dinv, xw2, agg2, E, H2);
  scatter_max_self_v4<<<cdiv((size_t)N * (H2 >> 2), 256), 256, 0, stream>>>(
      dinv, xw2, agg2, N, H2);
  bias_relu<<<cdiv(NF2, 256), 256, 0, stream>>>(agg2, b2, x2, NF2, H2);

  // ---- attention stage 2 (cat = [out(64) | g(256)] -> K=320) ----
  launch_gemm(x2, a2W, a2b, att,  hidx, NHB, H2, H2, 0, stream);
  softmax_rows<<<cdiv((size_t)NHB * 32, 256), 256, 0, stream>>>(att, NHB, H2);
  launch_gemm(x2, f2W, f2b, feat, hidx, NHB, H2, H2, 0, stream);
  attn_reduce<<<cdiv((size_t)Bn * H2, 256), 256, 0, stream>>>(att, feat, outb, Bn, NH, H2);
  concat2<<<cdiv((size_t)Bn * (H2 + H1), 256), 256, 0, stream>>>(outb, H2, g, H1, cat, Bn);
  launch_gemm(cat, g2W, g2b, gtmp, nullptr, Bn, H2 + H1, H1, 0, stream);
  add_inplace<<<cdiv((size_t)Bn * H1, 256), 256, 0, stream>>>(g, gtmp, (size_t)Bn * H1);

  // ---- head: out = relu(g@o1W + o1b) @ o2W + o2b ----
  launch_gemm(g, o1W, o1b, hbuf, nullptr, Bn, H1, H2, 1, stream);
  head_out<<<cdiv(Bn, 256), 256, 0, stream>>>(hbuf, o2W, o2b, (float*)d_out, Bn, H2);
}